// InputEmbeddings_82257213653584
// MI455X (gfx1250) — compile-verified
//
#include <hip/hip_runtime.h>

// out[b,s,d] = emb[x[b,s], d] + P[s, d]
// P[s, 2i]   = sin(s * 10000^(-2i/64)),  P[s, 2i+1] = cos(...)
//
// One thread owns (s, j): computes its 4 PE values ONCE (accurate sinf/cosf,
// since angles reach ~8191 rad — outside v_sin_f32's valid range), then streams
// BGROUP batch rows: idx load -> coalesced 256B row gather -> add -> NT store.

#define SEQ_LEN    8192
#define SEQ_SHIFT  13
#define EMB_D      64
#define BGROUP     8

typedef __attribute__((ext_vector_type(4))) float v4f;

__global__ __launch_bounds__(256) void emb_pe_gather_kernel(
    const int* __restrict__ x,
    const float* __restrict__ emb,
    float* __restrict__ out,
    int batch)
{
    const int tid = blockIdx.x * blockDim.x + threadIdx.x;
    const int j   = tid & 15;                 // v4f slot within 64-float row
    const int sb  = tid >> 4;
    const int s   = sb & (SEQ_LEN - 1);       // sequence position
    const int bg  = sb >> SEQ_SHIFT;          // batch group
    const int b0  = bg * BGROUP;
    if (b0 >= batch) return;

    // --- positional encoding, computed once per thread ---
    // freq_i = 10000^(-i/32) = exp2(-i * log2(10000)/32)
    const float c  = -0.41524101186092367f;   // -log2(10000)/32
    const float k  = (float)s;
    const float f0 = exp2f(c * (float)(2 * j));
    const float f1 = exp2f(c * (float)(2 * j + 1));
    const float a0 = k * f0;
    const float a1 = k * f1;
    v4f pe;
    pe.x = sinf(a0);   // accurate range-reduced sin/cos (args up to ~8191 rad)
    pe.y = cosf(a0);
    pe.z = sinf(a1);
    pe.w = cosf(a1);

    if (b0 + BGROUP <= batch) {
        // Fast path: fully unrolled -> 8 independent idx loads issued up front,
        // 8 in-flight b128 gathers, 8 NT b128 stores.
        #pragma unroll
        for (int bb = 0; bb < BGROUP; ++bb) {
            const int token = (b0 + bb) * SEQ_LEN + s;
            const int idx   = x[token];
            const v4f* __restrict__ src = (const v4f*)(emb + (size_t)idx * EMB_D);
            v4f v = src[j] + pe;
            v4f* __restrict__ dst = (v4f*)out + (size_t)token * (EMB_D / 4);
            __builtin_nontemporal_store(v, dst + j);
        }
    } else {
        for (int b = b0; b < batch; ++b) {
            const int token = b * SEQ_LEN + s;
            const int idx   = x[token];
            const v4f* __restrict__ src = (const v4f*)(emb + (size_t)idx * EMB_D);
            v4f v = src[j] + pe;
            v4f* __restrict__ dst = (v4f*)out + (size_t)token * (EMB_D / 4);
            __builtin_nontemporal_store(v, dst + j);
        }
    }
}

extern "C" void kernel_launch(void* const* d_in, const int* in_sizes, int n_in,
                              void* d_out, int out_size, void* d_ws, size_t ws_size,
                              hipStream_t stream)
{
    const int*   x   = (const int*)d_in[0];    // [32, 8192] token ids
    const float* emb = (const float*)d_in[1];  // [50257, 64] fp32
    float*       out = (float*)d_out;          // [32, 8192, 64] fp32

    const int n_tokens = in_sizes[0];          // 262144
    const int batch    = n_tokens / SEQ_LEN;   // 32
    const int bgroups  = (batch + BGROUP - 1) / BGROUP;

    const int threads  = SEQ_LEN * 16 * bgroups;  // 16 lanes per (s, bgroup)
    const int block    = 256;                     // 8 wave32 per block
    const int grid     = (threads + block - 1) / block;

    emb_pe_gather_kernel<<<grid, block, 0, stream>>>(x, emb, out, batch);
}